// PhysicsInformedLoss_83382495084544
// MI455X (gfx1250) — compile-verified
//
#include <hip/hip_runtime.h>
#include <hip/hip_bf16.h>
#include <stdint.h>

#define IDIM   512
#define NOUT   510
#define TILE   30          // 510 = 17 * 30, tiles exactly cover the domain
#define TPD    17
#define NBATCH 32
#define LDSP   33          // TDM-packed pitch of u/v/p tiles (33 cols)
#define LDSR   34          // +1 row to absorb edge-clamp shift
#define PNP    32          // p_next tile is 32x32

#define INV_DX  100.0f
#define INV_DX2 10000.0f
#define INV_DT  1000.0f
#define NU_C    1.0f       // VISC/DX = 0.01/0.01

typedef unsigned int v4u __attribute__((ext_vector_type(4)));
typedef int   v4i __attribute__((ext_vector_type(4)));
typedef int   v8i __attribute__((ext_vector_type(8)));
typedef float v2f __attribute__((ext_vector_type(2)));
typedef float v8f __attribute__((ext_vector_type(8)));

// One 2D TDM descriptor load: tile_h rows x tile_w f32 elements, row stride 512,
// packed contiguously into LDS at lds_off_bytes.
__device__ __forceinline__ void tdm_load_tile(uint32_t lds_off_bytes,
                                              const float* gsrc,
                                              int tile_w, int tile_h,
                                              int rem_w, int rem_h) {
  uint64_t ga = (uint64_t)(uintptr_t)gsrc;
  v4u g0;
  g0[0] = 1u;                                   // count=1, user descriptor
  g0[1] = lds_off_bytes;                        // lds_addr
  g0[2] = (uint32_t)ga;                         // global_addr[31:0]
  g0[3] = (uint32_t)((ga >> 32) & 0x01FFFFFFu)  // global_addr[56:32]
        | (2u << 30);                           // type=2 ("image")
  v8i g1;
  g1[0] = (2 << 16);                            // data_size=4B; no mask/flags
  g1[1] = (int)((rem_w & 0xFFFF) << 16);        // tensor_dim0[15:0]
  g1[2] = (int)(((rem_w >> 16) & 0xFFFF) | ((rem_h & 0xFFFF) << 16));
  g1[3] = (int)(((rem_h >> 16) & 0xFFFF) | ((tile_w & 0xFFFF) << 16));
  g1[4] = (int)(tile_h & 0xFFFF);               // tile_dim1; tile_dim2=0
  g1[5] = IDIM;                                 // tensor_dim0_stride = 512
  g1[6] = 0;
  g1[7] = 0;
  v4i gz4  = {0, 0, 0, 0};                      // 2D: groups 2/3 unused
  v8i gz8  = {0, 0, 0, 0, 0, 0, 0, 0};          // extra group (6-arg toolchain form)
  __builtin_amdgcn_tensor_load_to_lds(g0, g1, gz4, gz4, gz8, 0);
}

__global__ __launch_bounds__(256)
void pinn_main(const float* __restrict__ gen,
               const float* __restrict__ pnext,
               float* __restrict__ partials) {
  __shared__ __attribute__((aligned(16))) float su[LDSR * LDSP + 4];
  __shared__ __attribute__((aligned(16))) float sv[LDSR * LDSP + 4];
  __shared__ __attribute__((aligned(16))) float sp[LDSR * LDSP + 4];
  __shared__ __attribute__((aligned(16))) float spn[PNP * PNP];
  __shared__ float red[8];

  const int tidx = threadIdx.x;
  const int b    = blockIdx.y;
  const int ti   = blockIdx.x / TPD;
  const int tj   = blockIdx.x % TPD;
  const int gi0  = ti * TILE;
  const int gj0  = tj * TILE;

  const float* u  = gen   + (size_t)(b * 3 + 0) * IDIM * IDIM;
  const float* v  = gen   + (size_t)(b * 3 + 1) * IDIM * IDIM;
  const float* p  = gen   + (size_t)(b * 3 + 2) * IDIM * IDIM;
  const float* pn = pnext + (size_t)b * IDIM * IDIM;

  // Wave 0 issues 4 TDM DMAs (halo window rows gi0-1..gi0+31, cols gj0-1..gj0+31;
  // clamped at the 0-edge with an LDS shift so lds(row) = grow-gi0+1 uniformly).
  if (tidx < 32) {
    int r0 = gi0 - 1, rsh = 0; if (r0 < 0) { r0 = 0; rsh = 1; }
    int c0 = gj0 - 1, csh = 0; if (c0 < 0) { c0 = 0; csh = 1; }
    uint32_t shift = (uint32_t)(rsh * LDSP + csh) * 4u;
    size_t goff = (size_t)r0 * IDIM + c0;
    int remw = IDIM - c0, remh = IDIM - r0;
    tdm_load_tile((uint32_t)(uintptr_t)su + shift, u + goff, LDSP, LDSP, remw, remh);
    tdm_load_tile((uint32_t)(uintptr_t)sv + shift, v + goff, LDSP, LDSP, remw, remh);
    tdm_load_tile((uint32_t)(uintptr_t)sp + shift, p + goff, LDSP, LDSP, remw, remh);
    tdm_load_tile((uint32_t)(uintptr_t)spn, pn + (size_t)gi0 * IDIM + gj0,
                  PNP, PNP, IDIM - gj0, IDIM - gi0);
    __builtin_amdgcn_s_wait_tensorcnt(0);
  }
  __syncthreads();

  // LDS accessors: offsets (a,b) are relative to output point (I,J); LDS holds
  // global rows gi0-1.. so lds row = i + a + 1.
#define U_(a,c)  su[(i + (a) + 1) * LDSP + (j + (c) + 1)]
#define V_(a,c)  sv[(i + (a) + 1) * LDSP + (j + (c) + 1)]
#define P_(a,c)  sp[(i + (a) + 1) * LDSP + (j + (c) + 1)]
#define PN_(a,c) spn[(i + (a)) * PNP + (j + (c))]

  float acc = 0.0f;
  for (int o = tidx; o < TILE * TILE; o += 256) {
    int i = o / TILE, j = o % TILE;
    int I = gi0 + i, J = gj0 + j;

    // u-momentum fluxes: fe = avg_x(u)^2 - nu*ddx(u); fn = avg_x(v)*avg_y(u) - nu*ddy(u)
    auto fe = [&](int a, int c) {
      float m = 0.5f * (U_(a, c) + U_(a, c + 1));
      return m * m - NU_C * (U_(a, c + 1) - U_(a, c));
    };
    auto fn = [&](int a, int c) {
      return 0.25f * (V_(a, c) + V_(a, c + 1)) * (U_(a, c) + U_(a + 1, c))
             - NU_C * (U_(a + 1, c) - U_(a, c));
    };
    // v-momentum fluxes: ge = avg_y(u)*avg_x(v) - nu*ddx(v); gn = avg_y(v)^2 - nu*ddy(v)
    auto ge = [&](int a, int c) {
      return 0.25f * (U_(a, c) + U_(a + 1, c)) * (V_(a, c) + V_(a, c + 1))
             - NU_C * (V_(a, c + 1) - V_(a, c));
    };
    auto gn = [&](int a, int c) {
      float m = 0.5f * (V_(a, c) + V_(a + 1, c));
      return m * m - NU_C * (V_(a + 1, c) - V_(a, c));
    };

    float cont = (U_(1, 1) - U_(1, 0) + V_(1, 1) - V_(0, 1)) * INV_DX;

    float fe_11 = fe(1, 1), fe_10 = fe(1, 0), fe_1m = fe(1, -1);
    float fn_11 = fn(1, 1), fn_01 = fn(0, 1), fn_10 = fn(1, 0), fn_00 = fn(0, 0);
    float ge_11 = ge(1, 1), ge_10 = ge(1, 0), ge_01 = ge(0, 1), ge_00 = ge(0, 0);
    float gn_11 = gn(1, 1), gn_01 = gn(0, 1), gn_m1 = gn(-1, 1);

    float dudt_c = -(fe_11 - fe_10 + fn_11 - fn_01 + P_(1, 2) - P_(1, 1)) * INV_DX;
    float dudt_w = -(fe_10 - fe_1m + fn_10 - fn_00 + P_(1, 1) - P_(1, 0)) * INV_DX;
    float dvdt_c = -(ge_11 - ge_10 + gn_11 - gn_01 + P_(2, 1) - P_(1, 1)) * INV_DX;
    float dvdt_n = -(ge_01 - ge_00 + gn_01 - gn_m1 + P_(1, 1) - P_(0, 1)) * INV_DX;

    // padding masks from jnp.pad of dudt ((1,1),(1,2)) and dvdt ((1,2),(1,1))
    dudt_c = (J <= NOUT - 2) ? dudt_c : 0.0f;
    dudt_w = (J >= 1)        ? dudt_w : 0.0f;
    dvdt_c = (I <= NOUT - 2) ? dvdt_c : 0.0f;
    dvdt_n = (I >= 1)        ? dvdt_n : 0.0f;

    float b_conv = cont * INV_DT + (dudt_c - dudt_w + dvdt_c - dvdt_n) * INV_DX;

    float pp11 = PN_(1, 1) - P_(1, 1);
    float pp12 = PN_(1, 2) - P_(1, 2);
    float pp10 = PN_(1, 0) - P_(1, 0);
    float pp21 = PN_(2, 1) - P_(2, 1);
    float pp01 = PN_(0, 1) - P_(0, 1);
    float lap  = 4.0f * pp11 - pp12 - pp10 - pp21 - pp01;
    float pres = lap * INV_DX2 + b_conv;

    acc += fabsf(cont) + fabsf(pres);
  }
#undef U_
#undef V_
#undef P_
#undef PN_

  // Wave-level reduction via V_WMMA_F32_16X16X4_F32 with all-ones B:
  // A[m,0]=acc_m (lanes 0-15), A[m,2]=acc_{m+16} -> D[m,n] = acc_m + acc_{m+16}.
  v2f A;  A[0] = acc;  A[1] = 0.0f;
  v2f Bv; Bv[0] = 1.0f; Bv[1] = 1.0f;
  v8f C = {0.0f, 0.0f, 0.0f, 0.0f, 0.0f, 0.0f, 0.0f, 0.0f};
  C = __builtin_amdgcn_wmma_f32_16x16x4_f32(false, A, false, Bv, (short)0, C,
                                            false, false);
  float s = C[0] + C[1] + C[2] + C[3] + C[4] + C[5] + C[6] + C[7];
  s += __shfl_xor(s, 16, 32);   // lanes n / n+16 hold complementary half-sums

  int lane = tidx & 31, wid = tidx >> 5;
  if (lane == 0) red[wid] = s;
  __syncthreads();
  if (tidx == 0) {
    float t = 0.0f;
#pragma unroll
    for (int w = 0; w < 8; ++w) t += red[w];
    partials[(size_t)b * (TPD * TPD) + blockIdx.x] = t;
  }
}

__global__ __launch_bounds__(256)
void pinn_bc(const float* __restrict__ gen, float* __restrict__ bcout) {
  __shared__ float r0[256], r1[256], r2[256], r3[256];
  const int b = blockIdx.x, t = threadIdx.x;
  const float* u = gen + (size_t)(b * 3 + 0) * IDIM * IDIM;
  const float* v = gen + (size_t)(b * 3 + 1) * IDIM * IDIM;
  const float* p = gen + (size_t)(b * 3 + 2) * IDIM * IDIM;

  float y0 = 0.f, yL = 0.f, x0 = 0.f, xL = 0.f;
  for (int j = 1 + t; j <= 509; j += 256) {
    y0 += u[j] + u[IDIM + j];
    yL -= u[510 * IDIM + j] + u[511 * IDIM + j];
  }
  for (int j = 1 + t; j <= 510; j += 256) {
    y0 += v[j] + p[j];
    yL += v[511 * IDIM + j] + p[511 * IDIM + j];
  }
  for (int i = 1 + t; i <= 509; i += 256) {
    x0 += v[i * IDIM + 0] + v[i * IDIM + 1];
    xL += v[i * IDIM + 511] + v[i * IDIM + 510];
  }
  for (int i = 1 + t; i <= 510; i += 256) {
    x0 += u[i * IDIM + 0] + p[i * IDIM + 0];
    xL += u[i * IDIM + 511] + p[i * IDIM + 511];
  }
  r0[t] = y0; r1[t] = yL; r2[t] = x0; r3[t] = xL;
  __syncthreads();
  for (int st = 128; st > 0; st >>= 1) {
    if (t < st) {
      r0[t] += r0[t + st]; r1[t] += r1[t + st];
      r2[t] += r2[t + st]; r3[t] += r3[t + st];
    }
    __syncthreads();
  }
  if (t == 0) {
    float yLt = r1[0] + 2.0f * 1.0f * 509.0f;   // 509 copies of 2*U0
    bcout[b] = fabsf(r0[0]) + fabsf(yLt) + fabsf(r2[0]) + fabsf(r3[0]);
  }
}

__global__ __launch_bounds__(256)
void pinn_final(const float* __restrict__ partials,
                const float* __restrict__ bc,
                float* __restrict__ out) {
  __shared__ float sr[256];
  const int t = threadIdx.x;
  float s = 0.0f;
  for (int k = t; k < NBATCH * TPD * TPD; k += 256) s += partials[k];
  float sb = (t < NBATCH) ? bc[t] : 0.0f;

  sr[t] = s;
  __syncthreads();
  for (int st = 128; st > 0; st >>= 1) {
    if (t < st) sr[t] += sr[t + st];
    __syncthreads();
  }
  float stot = sr[0];
  __syncthreads();
  sr[t] = sb;
  __syncthreads();
  for (int st = 128; st > 0; st >>= 1) {
    if (t < st) sr[t] += sr[t + st];
    __syncthreads();
  }
  if (t == 0) {
    // 0.4*(mean cont + mean poisson averaged over batch) + 0.2*sum(bc)
    out[0] = 0.4f * stot / (float)((size_t)NBATCH * NOUT * NOUT) + 0.2f * sr[0];
  }
}

extern "C" void kernel_launch(void* const* d_in, const int* in_sizes, int n_in,
                              void* d_out, int out_size, void* d_ws, size_t ws_size,
                              hipStream_t stream) {
  (void)in_sizes; (void)n_in; (void)out_size; (void)ws_size;
  const float* gen = (const float*)d_in[0];
  const float* pnx = (const float*)d_in[1];
  float* out = (float*)d_out;
  float* partials = (float*)d_ws;                       // 32*17*17 = 9248 floats
  float* bc = partials + NBATCH * TPD * TPD;            // +32 floats

  dim3 grid(TPD * TPD, NBATCH);
  pinn_main<<<grid, 256, 0, stream>>>(gen, pnx, partials);
  pinn_bc<<<NBATCH, 256, 0, stream>>>(gen, bc);
  pinn_final<<<1, 256, 0, stream>>>(partials, bc, out);
}